// MaskPruningGlobalAttentionChannel_74887049773155
// MI455X (gfx1250) — compile-verified
//
#include <hip/hip_runtime.h>

typedef __attribute__((ext_vector_type(2))) float v2f;
typedef __attribute__((ext_vector_type(8))) float v8f;

#define BB 8
#define CC 256
#define HWN 4096

// -----------------------------------------------------------------------------
// Generic batched GEMM: D[b] = A[b] (256x256) @ Bm[b] (256x4096)  (+ bias[row])
// One wave computes a 16(M) x 64(N) strip with V_WMMA_F32_16X16X4_F32.
// Block = 128 threads = 4 waves stacked along M (covers 64x64 per block).
// -----------------------------------------------------------------------------
__global__ __launch_bounds__(128) void gemm256_kernel(
    const float* __restrict__ A, long aStride,
    const float* __restrict__ bias,
    const float* __restrict__ Bm, long bStride,
    float* __restrict__ D, long dStride)
{
    const int lane  = threadIdx.x & 31;
    const int wave  = threadIdx.x >> 5;
    const int g     = lane >> 4;      // half-wave group (0/1)
    const int ln    = lane & 15;
    const int batch = blockIdx.z;
    const int mBase = blockIdx.y * 64 + wave * 16;
    const int nBase = blockIdx.x * 64;

    const float* Ab = A + (long)batch * aStride;   // [256 x 256]
    const float* Bb = Bm + (long)batch * bStride;  // [256 x 4096]
    float*       Db = D + (long)batch * dStride;   // [256 x 4096]

    v8f acc0 = {}, acc1 = {}, acc2 = {}, acc3 = {};
    const int m  = mBase + ln;
    const int n0 = nBase + ln;

    #pragma unroll 4
    for (int k = 0; k < CC; k += 4) {
        // A fragment (16x4): a.x = A[m, k+2g], a.y = A[m, k+2g+1]  (contiguous)
        v2f a = *(const v2f*)(Ab + (long)m * CC + k + 2 * g);
        // B fragments (4x16): b.x = B[k+2g, n], b.y = B[k+2g+1, n]
        const float* br0 = Bb + (long)(k + 2 * g) * HWN;
        const float* br1 = br0 + HWN;
        v2f b0 = { br0[n0     ], br1[n0     ] };
        v2f b1 = { br0[n0 + 16], br1[n0 + 16] };
        v2f b2 = { br0[n0 + 32], br1[n0 + 32] };
        v2f b3 = { br0[n0 + 48], br1[n0 + 48] };
        acc0 = __builtin_amdgcn_wmma_f32_16x16x4_f32(false, a, false, b0, (short)0, acc0, false, false);
        acc1 = __builtin_amdgcn_wmma_f32_16x16x4_f32(false, a, false, b1, (short)0, acc1, false, false);
        acc2 = __builtin_amdgcn_wmma_f32_16x16x4_f32(false, a, false, b2, (short)0, acc2, false, false);
        acc3 = __builtin_amdgcn_wmma_f32_16x16x4_f32(false, a, false, b3, (short)0, acc3, false, false);
    }

    // Epilogue: VGPR r holds row M = r + 8*g, column N = lane&15.
    #pragma unroll
    for (int r = 0; r < 8; ++r) {
        const int mo = mBase + r + 8 * g;
        const float bv = bias ? bias[mo] : 0.0f;
        float* drow = Db + (long)mo * HWN + n0;
        drow[0]  = acc0[r] + bv;
        drow[16] = acc1[r] + bv;
        drow[32] = acc2[r] + bv;
        drow[48] = acc3[r] + bv;
    }
}

// -----------------------------------------------------------------------------
// corr[b] = Q[b] (256x4096) @ K[b]^T (4096x256)  -> [256x256]
// Both fragments are contiguous float2 loads along the K (=HW) axis.
// -----------------------------------------------------------------------------
__global__ __launch_bounds__(128) void corr_kernel(
    const float* __restrict__ Q, const float* __restrict__ Km,
    float* __restrict__ corr)
{
    const int lane  = threadIdx.x & 31;
    const int wave  = threadIdx.x >> 5;
    const int g     = lane >> 4;
    const int ln    = lane & 15;
    const int batch = blockIdx.z;
    const int mBase = blockIdx.y * 64 + wave * 16;
    const int nBase = blockIdx.x * 16;

    const float* qp = Q  + (long)batch * CC * HWN + (long)(mBase + ln) * HWN;
    const float* kp = Km + (long)batch * CC * HWN + (long)(nBase + ln) * HWN;

    v8f acc = {};
    #pragma unroll 8
    for (int k = 0; k < HWN; k += 4) {
        v2f a = *(const v2f*)(qp + k + 2 * g);   // Q[m, k+2g : k+2g+2]
        v2f b = *(const v2f*)(kp + k + 2 * g);   // K[n, k+2g : k+2g+2]  (== B^T frag)
        acc = __builtin_amdgcn_wmma_f32_16x16x4_f32(false, a, false, b, (short)0, acc, false, false);
    }

    float* cb = corr + (long)batch * CC * CC;
    #pragma unroll
    for (int r = 0; r < 8; ++r) {
        cb[(long)(mBase + r + 8 * g) * CC + nBase + ln] = acc[r];
    }
}

// -----------------------------------------------------------------------------
// Fused row softmax (over HW=4096) + attended/v product + final blend.
// One 256-thread block per (b,c) row; scores live in d_out and are
// overwritten in place (each thread writes exactly the indices it read).
// -----------------------------------------------------------------------------
__global__ __launch_bounds__(256) void softmax_out_kernel(
    const float* __restrict__ scores,
    const float* __restrict__ fg, const float* __restrict__ mask,
    const float* __restrict__ V,  const float* __restrict__ gamma,
    float* __restrict__ out)
{
    const int  tid  = threadIdx.x;
    const int  lane = tid & 31;
    const int  wid  = tid >> 5;
    const long base = (long)blockIdx.x * HWN;

    __shared__ float red[8];

    float vals[16];
    float mx = -3.402823466e38f;
    #pragma unroll
    for (int t = 0; t < 16; ++t) {
        vals[t] = scores[base + t * 256 + tid];
        mx = fmaxf(mx, vals[t]);
    }
    #pragma unroll
    for (int off = 16; off >= 1; off >>= 1)
        mx = fmaxf(mx, __shfl_xor(mx, off, 32));
    if (lane == 0) red[wid] = mx;
    __syncthreads();
    #pragma unroll
    for (int w = 0; w < 8; ++w) mx = fmaxf(mx, red[w]);
    __syncthreads();

    float sum = 0.0f;
    #pragma unroll
    for (int t = 0; t < 16; ++t) {
        vals[t] = __expf(vals[t] - mx);
        sum += vals[t];
    }
    #pragma unroll
    for (int off = 16; off >= 1; off >>= 1)
        sum += __shfl_xor(sum, off, 32);
    if (lane == 0) red[wid] = sum;
    __syncthreads();
    float tot = 0.0f;
    #pragma unroll
    for (int w = 0; w < 8; ++w) tot += red[w];

    const float inv = 1.0f / tot;
    const float gam = gamma[0];
    #pragma unroll
    for (int t = 0; t < 16; ++t) {
        const long i = base + t * 256 + tid;
        const float att = vals[t] * inv * V[i];
        const float mk  = mask[i];
        out[i] = fg[i] * mk + gam * (1.0f - mk) * att;
    }
}

// -----------------------------------------------------------------------------
extern "C" void kernel_launch(void* const* d_in, const int* in_sizes, int n_in,
                              void* d_out, int out_size, void* d_ws, size_t ws_size,
                              hipStream_t stream)
{
    (void)in_sizes; (void)n_in; (void)out_size; (void)ws_size;

    const float* fg    = (const float*)d_in[0];
    const float* bg    = (const float*)d_in[1];
    const float* mask  = (const float*)d_in[2];
    const float* Wq    = (const float*)d_in[3];
    const float* bq    = (const float*)d_in[4];
    const float* Wk    = (const float*)d_in[5];
    const float* bk    = (const float*)d_in[6];
    const float* Wv    = (const float*)d_in[7];
    const float* bv    = (const float*)d_in[8];
    const float* gamma = (const float*)d_in[9];
    float* out = (float*)d_out;

    const long mat = (long)BB * CC * HWN;      // elements per [B,C,HW] tensor
    float* Q    = (float*)d_ws;
    float* Km   = Q  + mat;
    float* Vv   = Km + mat;
    float* corr = Vv + mat;                    // [B, C, C]

    const dim3 blk(128);
    const dim3 gGemm(HWN / 64, CC / 64, BB);
    const long xStride = (long)CC * HWN;

    // q/k/v 1x1 convs (weight shared across batch => aStride = 0)
    gemm256_kernel<<<gGemm, blk, 0, stream>>>(Wq, 0, bq, fg, xStride, Q,  xStride);
    gemm256_kernel<<<gGemm, blk, 0, stream>>>(Wk, 0, bk, bg, xStride, Km, xStride);
    gemm256_kernel<<<gGemm, blk, 0, stream>>>(Wv, 0, bv, fg, xStride, Vv, xStride);

    // corr = q k^T
    const dim3 gCorr(CC / 16, CC / 64, BB);
    corr_kernel<<<gCorr, blk, 0, stream>>>(Q, Km, corr);

    // scores = corr @ mask  -> staged in d_out
    gemm256_kernel<<<gGemm, blk, 0, stream>>>(corr, (long)CC * CC, nullptr,
                                              mask, xStride, out, xStride);

    // softmax + blend, in place on d_out
    softmax_out_kernel<<<dim3(BB * CC), dim3(256), 0, stream>>>(
        out, fg, mask, Vv, gamma, out);
}